// ModulatedDeformConv_blur_73418170958219
// MI455X (gfx1250) — compile-verified
//
#include <hip/hip_runtime.h>
#include <hip/hip_bf16.h>

typedef __attribute__((ext_vector_type(2))) float v2f;
typedef __attribute__((ext_vector_type(8))) float v8f;

#define HH 256
#define WW 256
#define KK 81
#define PLANE (HH * WW)

// One wave = 16 consecutive output pixels of one (b, y) row.
// k-contraction via V_WMMA_F32_16X16X4_F32 (full fp32):
//   A (16x4)  : w[k] replicated down M; lane half selects K={0,1} vs K={2,3}
//   B (4x16)  : val[k, pixel] (mask * bilinear sample), same K striping
//   C/D(16x16): row M=0 (VGPR0, lanes 0-15) = sum_k w[k]*val[k,n]
__device__ __forceinline__ void dcn_sample(
    const float* __restrict__ g0, const float* __restrict__ g1,
    const float* __restrict__ g2, const float* __restrict__ offB,
    const float* __restrict__ mskB, int k, float fy, float fx,
    float& v0, float& v1, float& v2)
{
    const int   ky = k / 9;
    const int   kx = k - ky * 9;
    // streaming, read-once data -> non-temporal so input gathers keep cache
    const float dy = __builtin_nontemporal_load(offB + (size_t)(2 * k) * PLANE);
    const float dx = __builtin_nontemporal_load(offB + (size_t)(2 * k + 1) * PLANE);
    const float m  = __builtin_nontemporal_load(mskB + (size_t)k * PLANE);
    const float py = fy + (float)ky + dy;
    const float px = fx + (float)kx + dx;
    const float y0f = floorf(py), x0f = floorf(px);
    const float wy = py - y0f,   wx = px - x0f;
    const int y0 = (int)y0f, x0 = (int)x0f;
    const int y1 = y0 + 1,   x1 = x0 + 1;
    const float vy0 = (y0 >= 0 && y0 < HH) ? 1.f : 0.f;
    const float vy1 = (y1 >= 0 && y1 < HH) ? 1.f : 0.f;
    const float vx0 = (x0 >= 0 && x0 < WW) ? 1.f : 0.f;
    const float vx1 = (x1 >= 0 && x1 < WW) ? 1.f : 0.f;
    const int yc0 = min(max(y0, 0), HH - 1);
    const int yc1 = min(max(y1, 0), HH - 1);
    const int xc0 = min(max(x0, 0), WW - 1);
    const int xc1 = min(max(x1, 0), WW - 1);
    const int i00 = yc0 * WW + xc0;
    const int i01 = yc0 * WW + xc1;
    const int i10 = yc1 * WW + xc0;
    const int i11 = yc1 * WW + xc1;
    const float w00 = (1.f - wy) * (1.f - wx) * vy0 * vx0 * m;
    const float w01 = (1.f - wy) * wx         * vy0 * vx1 * m;
    const float w10 = wy         * (1.f - wx) * vy1 * vx0 * m;
    const float w11 = wy         * wx         * vy1 * vx1 * m;
    v0 = g0[i00]*w00 + g0[i01]*w01 + g0[i10]*w10 + g0[i11]*w11;
    v1 = g1[i00]*w00 + g1[i01]*w01 + g1[i10]*w10 + g1[i11]*w11;
    v2 = g2[i00]*w00 + g2[i01]*w01 + g2[i10]*w10 + g2[i11]*w11;
}

__global__ __launch_bounds__(256)
void dcn_wmma_kernel(const float* __restrict__ input,
                     const float* __restrict__ offset,
                     const float* __restrict__ mask,
                     const float* __restrict__ weight,
                     const float* __restrict__ bias,
                     float* __restrict__ out)
{
    const int lane  = threadIdx.x & 31;
    const int wave  = blockIdx.x * 8 + (threadIdx.x >> 5);   // 8 waves / block
    const int xtile = wave & 15;            // 16 tiles per row
    const int y     = (wave >> 4) & 255;    // Ho = 256
    const int b     = wave >> 12;           // B = 2
    const int p     = lane & 15;            // pixel within tile
    const int half  = lane >> 4;            // K-pair selector per WMMA layout
    const int x     = xtile * 16 + p;

    const float* g0 = input + (size_t)b * 3 * PLANE;
    const float* g1 = g0 + PLANE;
    const float* g2 = g0 + 2 * PLANE;
    const size_t pos = (size_t)y * WW + x;
    const float* offB = offset + (size_t)b * 2 * KK * PLANE + pos;
    const float* mskB = mask   + (size_t)b * KK * PLANE + pos;

    v8f acc0 = {}; v8f acc1 = {}; v8f acc2 = {};
    const float fy = (float)(y - 4);
    const float fx = (float)(x - 4);

    // ---- 20 full K=4 chunks: k = 0..79, branch-free hot loop ----
    for (int kc = 0; kc < 20; ++kc) {
        v2f a, b0, b1, b2;
#pragma unroll
        for (int j = 0; j < 2; ++j) {
            const int k = kc * 4 + half * 2 + j;   // always < 80
            float v0, v1, v2;
            dcn_sample(g0, g1, g2, offB, mskB, k, fy, fx, v0, v1, v2);
            const float w = weight[k];
            if (j == 0) { a.x = w; b0.x = v0; b1.x = v1; b2.x = v2; }
            else        { a.y = w; b0.y = v0; b1.y = v1; b2.y = v2; }
        }
        acc0 = __builtin_amdgcn_wmma_f32_16x16x4_f32(false, a, false, b0, (short)0, acc0, false, false);
        acc1 = __builtin_amdgcn_wmma_f32_16x16x4_f32(false, a, false, b1, (short)0, acc1, false, false);
        acc2 = __builtin_amdgcn_wmma_f32_16x16x4_f32(false, a, false, b2, (short)0, acc2, false, false);
    }

    // ---- tail chunk: only k=80 is live (half==0, j==0); rest weighted 0 ----
    {
        v2f a = {0.f, 0.f}, b0 = {0.f, 0.f}, b1 = {0.f, 0.f}, b2 = {0.f, 0.f};
        if (half == 0) {
            float v0, v1, v2;
            dcn_sample(g0, g1, g2, offB, mskB, KK - 1, fy, fx, v0, v1, v2);
            a.x = weight[KK - 1];
            b0.x = v0; b1.x = v1; b2.x = v2;
        }
        acc0 = __builtin_amdgcn_wmma_f32_16x16x4_f32(false, a, false, b0, (short)0, acc0, false, false);
        acc1 = __builtin_amdgcn_wmma_f32_16x16x4_f32(false, a, false, b1, (short)0, acc1, false, false);
        acc2 = __builtin_amdgcn_wmma_f32_16x16x4_f32(false, a, false, b2, (short)0, acc2, false, false);
    }

    const float bz = bias[0];
    if (lane < 16) {                 // D row M=0 lives in VGPR0, lanes 0-15
        float* o = out + ((size_t)(b * 3) * HH + y) * WW + x;
        o[0]         = acc0[0] + bz;
        o[PLANE]     = acc1[0] + bz;
        o[2 * PLANE] = acc2[0] + bz;
    }
}

extern "C" void kernel_launch(void* const* d_in, const int* in_sizes, int n_in,
                              void* d_out, int out_size, void* d_ws, size_t ws_size,
                              hipStream_t stream) {
    const float* input  = (const float*)d_in[0];
    const float* offset = (const float*)d_in[1];
    const float* mask   = (const float*)d_in[2];
    const float* weight = (const float*)d_in[3];
    const float* bias   = (const float*)d_in[4];
    float* out = (float*)d_out;
    // total waves = B * Ho * (Wo/16) = 2*256*16 = 8192 ; 8 waves/block -> 1024 blocks
    dcn_wmma_kernel<<<dim3(1024), dim3(256), 0, stream>>>(input, offset, mask,
                                                          weight, bias, out);
}